// FieldSampler_25331717112457
// MI455X (gfx1250) — compile-verified
//
#include <hip/hip_runtime.h>

// FieldSampler: B=8, G=4096, N=4096, D=256 (fixed by the reference setup).
// out[b,n,:] = lerp(field[b,il,:], field[b,il+1,:]) with il from a binary
// search of sample position in the sorted grid row.

#define B_ 8
#define G_ 4096
#define N_ 4096
#define D_ 256

#define SAMPLES_PER_BLOCK 4
#define THREADS_PER_SAMPLE 64   // 64 lanes * float4 = 256 = D
#define THREADS (SAMPLES_PER_BLOCK * THREADS_PER_SAMPLE)
#define BLOCKS_PER_BATCH (N_ / SAMPLES_PER_BLOCK)  // 1024

typedef float v4f __attribute__((ext_vector_type(4)));

__global__ __launch_bounds__(THREADS) void FieldSampler_kernel(
    const float* __restrict__ field,   // [B, G, D]
    const float* __restrict__ grid,    // [B, G]
    const float* __restrict__ pos,     // [B, N]
    float* __restrict__ out)           // [B, N, D]
{
    __shared__ int   s_idx[SAMPLES_PER_BLOCK];
    __shared__ float s_wl [SAMPLES_PER_BLOCK];
    __shared__ float s_wr [SAMPLES_PER_BLOCK];

    const int tid = threadIdx.x;
    // 1024 blocks per batch: cheap (b, n) decomposition, no 64-bit div.
    const int b  = blockIdx.x >> 10;                 // blockIdx / 1024
    const int n0 = (blockIdx.x & (BLOCKS_PER_BATCH - 1)) * SAMPLES_PER_BLOCK;

    // ---- Phase 1: one lane per sample does the binary search ----
    if (tid < SAMPLES_PER_BLOCK) {
        const int n = n0 + tid;
        const float* __restrict__ g = grid + (size_t)b * G_;

        float v = pos[(size_t)b * N_ + n];
        v = fminf(fmaxf(v, g[0]), g[G_ - 1]);

        // Branchless bit-filling upper_bound over g[0 .. G-2] (4095 = 2^12-1
        // elements). Invariant: ub only has bits > step set, so
        // ub + step - 1 <= 4094 -- every probe is in range, no guards.
        // Since g[0] <= v, ub >= 1, and il = ub-1 lands in [0, G-2],
        // matching clip(searchsorted_right(g, v) - 1, 0, G-2).
        int ub = 0;
        #pragma unroll
        for (int step = G_ / 2; step > 0; step >>= 1) {
            if (g[ub + step - 1] <= v) ub += step;   // load + cmp + cndmask
        }
        const int il = ub - 1;

        const float gl = g[il];
        const float gr = g[il + 1];
        const float denom = fmaxf(gr - gl, 1e-8f);
        const float wr = (v - gl) / denom;

        s_idx[tid] = il;
        s_wr[tid]  = wr;
        s_wl[tid]  = 1.0f - wr;

        // Warm L2/L0 for the two gathered rows (2 * 1KB = 16 cachelines)
        // while the rest of the block sits at the barrier (global_prefetch_b8).
        const float* fl = field + ((size_t)b * G_ + il) * D_;
        #pragma unroll
        for (int c = 0; c < 8; ++c) {
            __builtin_prefetch(fl + c * 32, 0, 3);        // row il
            __builtin_prefetch(fl + D_ + c * 32, 0, 3);   // row il+1
        }
    }
    __syncthreads();

    // ---- Phase 2: 64 lanes per sample do the vectorized lerp ----
    const int grp  = tid >> 6;   // sample index within block
    const int lane = tid & 63;   // float4 chunk within the D=256 row
    const int n    = n0 + grp;

    const int   il = s_idx[grp];
    const float wl = s_wl[grp];
    const float wr = s_wr[grp];

    const v4f* __restrict__ fl =
        (const v4f*)(field + ((size_t)b * G_ + il) * D_);
    const v4f* __restrict__ fr =
        (const v4f*)(field + ((size_t)b * G_ + (il + 1)) * D_);
    v4f* o = (v4f*)(out + ((size_t)b * N_ + n) * D_);

    const v4f a = fl[lane];          // global_load_b128 (field stays L2-resident)
    const v4f c = fr[lane];          // global_load_b128

    v4f r;
    r.x = wl * a.x + wr * c.x;
    r.y = wl * a.y + wr * c.y;
    r.z = wl * a.z + wr * c.z;
    r.w = wl * a.w + wr * c.w;

    // Streaming output: non-temporal so the 32MB result doesn't evict the
    // L2-resident field.
    __builtin_nontemporal_store(r, &o[lane]);
}

extern "C" void kernel_launch(void* const* d_in, const int* in_sizes, int n_in,
                              void* d_out, int out_size, void* d_ws, size_t ws_size,
                              hipStream_t stream) {
    const float* field = (const float*)d_in[0];  // [B, G, D]
    const float* grid  = (const float*)d_in[1];  // [B, G, 1]
    const float* pos   = (const float*)d_in[2];  // [B, N, 1]
    float* out = (float*)d_out;                  // [B, N, D]

    const int blocks = (B_ * N_) / SAMPLES_PER_BLOCK;  // 8192
    FieldSampler_kernel<<<blocks, THREADS, 0, stream>>>(field, grid, pos, out);
}